// Pooling_56358560858519
// MI455X (gfx1250) — compile-verified
//
#include <hip/hip_runtime.h>

// 2x2 non-overlapping max-pool, X:(16,64,256,256) f32 -> Y:(16,64,128,128) f32.
// Pure streaming problem: ~320 MiB of traffic, ~14 us floor at 23.3 TB/s.
// Strategy: one thread -> 4 consecutive output pixels (one b128 store),
// fed by 4 non-temporal b128 loads (8 floats x 2 input rows).
// A wave32 with a common output row covers one full 1024B input row with
// perfectly interleaved 16B accesses -> every HBM cacheline touched once.

typedef __attribute__((ext_vector_type(4))) float v4f;

#define IN_H   256
#define IN_W   256
#define OUT_H  128
#define OUT_W  128

__global__ __launch_bounds__(256, 8)
void maxpool2x2_nt_kernel(const float* __restrict__ X,
                          float* __restrict__ Y,
                          unsigned n_quads) {
    const unsigned idx = blockIdx.x * blockDim.x + threadIdx.x;
    if (idx >= n_quads) return;

    // idx -> (output row r over B*C*OUT_H, quad g of 4 output cols)
    const unsigned r  = idx >> 5;    // OUT_W/4 = 32 quads per output row
    const unsigned g  = idx & 31;
    const unsigned bc = r >> 7;      // OUT_H = 128
    const unsigned ho = r & 127;

    // input base: image bc, rows 2*ho and 2*ho+1, cols starting at g*8
    const size_t in0 = ((size_t)bc * IN_H + (size_t)ho * 2) * IN_W + (size_t)g * 8;
    const v4f* p0 = (const v4f*)(X + in0);          // row 2*ho   : floats [g*8, g*8+8)
    const v4f* p1 = (const v4f*)(X + in0 + IN_W);   // row 2*ho+1 : floats [g*8, g*8+8)

    // Non-temporal: streamed once, working set (256 MiB) > L2 (192 MB).
    const v4f a0 = __builtin_nontemporal_load(p0 + 0);
    const v4f a1 = __builtin_nontemporal_load(p0 + 1);
    const v4f b0 = __builtin_nontemporal_load(p1 + 0);
    const v4f b1 = __builtin_nontemporal_load(p1 + 1);

    // 4 output pixels; fmaxf trees -> v_max3_num_f32 + v_max_num_f32
    v4f o;
    o.x = fmaxf(fmaxf(a0.x, a0.y), fmaxf(b0.x, b0.y));
    o.y = fmaxf(fmaxf(a0.z, a0.w), fmaxf(b0.z, b0.w));
    o.z = fmaxf(fmaxf(a1.x, a1.y), fmaxf(b1.x, b1.y));
    o.w = fmaxf(fmaxf(a1.z, a1.w), fmaxf(b1.z, b1.w));

    __builtin_nontemporal_store(o, (v4f*)(Y + (size_t)idx * 4));
}

extern "C" void kernel_launch(void* const* d_in, const int* in_sizes, int n_in,
                              void* d_out, int out_size, void* d_ws, size_t ws_size,
                              hipStream_t stream) {
    (void)in_sizes; (void)n_in; (void)d_ws; (void)ws_size;
    const float* X = (const float*)d_in[0];
    float* Y = (float*)d_out;

    const unsigned n_quads = (unsigned)(out_size / 4);   // 4 outputs per thread
    const unsigned block = 256;                          // 8 wave32s
    const unsigned grid  = (n_quads + block - 1) / block;

    maxpool2x2_nt_kernel<<<grid, block, 0, stream>>>(X, Y, n_quads);
}